// SpikingLayer_81346680586499
// MI455X (gfx1250) — compile-verified
//
#include <hip/hip_runtime.h>
#include <stdint.h>

// IAF spiking layer, MI455X (gfx1250), wave32.
//
// Roofline: 512 MB total traffic @ 23.3 TB/s => ~22 us floor; ~0.25 flop/byte,
// purely bandwidth bound. The spike-reset recurrence is a data-dependent
// sequential scan over T -> WMMA inapplicable. CDNA5 leverage: deep async
// global->LDS pipeline (ASYNCcnt), SADDR-form addressing, NT stores.
//
// Config: 128-thread blocks (4 wave32s), 4 floats/thread, 256 blocks.
// Pipeline: 3 LDS buffers x 16 timesteps -> 32 in-flight b128/thread steady
// state (~16 MB chip-wide in flight), max 48 outstanding/wave (< 63 cap).
// LDS: 3*16*128*16B = 96 KB/block -> ~3 blocks/WGP (320 KB).

typedef __attribute__((ext_vector_type(4))) float v4f;

constexpr int   kT      = 512;           // timesteps
constexpr int   kF      = 8192;          // features
constexpr int   kBlock  = 128;           // threads per block (4 wave32s)
constexpr int   kVec    = 4;             // floats per thread per step (b128)
constexpr int   kFTile  = kBlock * kVec; // 512 features per block
constexpr int   kNFBlk  = kF / kFTile;   // 16 feature tiles
constexpr int   kTChunk = 16;            // timesteps per async chunk
constexpr int   kNBuf   = 3;             // LDS pipeline depth
constexpr int   kNChunk = kT / kTChunk;  // 32 chunks

static_assert(kNChunk > kNBuf, "pipeline peel assumes more chunks than buffers");
static_assert((kNBuf)*kTChunk <= 63, "ASYNCcnt is 6 bits");

__global__ __launch_bounds__(kBlock) void iaf_spike_kernel(
    const float* __restrict__ in, float* __restrict__ out) {
  __shared__ v4f smem[kNBuf][kTChunk][kBlock];   // 96 KB

  const int tid  = threadIdx.x;
  const int fblk = blockIdx.x % kNFBlk;
  const int b    = blockIdx.x / kNFBlk;

  // Per-lane byte offset of this thread's 16B slice within one timestep row.
  const unsigned fbyte = (unsigned)((fblk * kFTile + tid * kVec) * 4);

  // Block-uniform base of this batch's [T, F] slab (kept in SGPRs via "s").
  const float* sbase = in  + (long long)b * kT * kF;
  float*       obase = out + (long long)b * kT * kF + (fbyte >> 2);

  // Precompute the 16 per-timestep global byte offsets once (vaddr operands).
  unsigned voff[kTChunk];
#pragma unroll
  for (int tt = 0; tt < kTChunk; ++tt)
    voff[tt] = fbyte + (unsigned)(tt * kF * 4);

  // Issue one chunk: kTChunk async b128 copies, SADDR form:
  //   global_load_async_to_lds_b128 vdst(lds addr), vaddr(u32 byte off), s[base]
  // Per-chunk base advances uniformly in SGPRs; zero per-issue VALU.
  auto issue_chunk = [&](int c) {
    const int    buf = c % kNBuf;
    const float* g   = sbase + (long long)c * kTChunk * kF;  // uniform
#pragma unroll
    for (int tt = 0; tt < kTChunk; ++tt) {
      // Flat pointer to __shared__ carries the LDS byte offset in its low 32b.
      unsigned laddr = (unsigned)(unsigned long long)(&smem[buf][tt][tid]);
      asm volatile("global_load_async_to_lds_b128 %0, %1, %2"
                   :: "v"(laddr), "v"(voff[tt]), "s"(g)
                   : "memory");
    }
  };

  // Single accumulator m = vmem - sub:  m += x; s = (m >= 1); m -= s.
  v4f m = {0.0f, 0.0f, 0.0f, 0.0f};

  auto process_chunk = [&](int c) {
    const int buf = c % kNBuf;
    float*    o   = obase + (long long)c * kTChunk * kF;
#pragma unroll
    for (int tt = 0; tt < kTChunk; ++tt) {
      v4f x = smem[buf][tt][tid];       // ds_load_b128
      m += x;                            // membrane after integration
      v4f s;
      s.x = (m.x >= 1.0f) ? 1.0f : 0.0f; // v_cmp + v_cndmask (inline 1.0)
      s.y = (m.y >= 1.0f) ? 1.0f : 0.0f;
      s.z = (m.z >= 1.0f) ? 1.0f : 0.0f;
      s.w = (m.w >= 1.0f) ? 1.0f : 0.0f;
      m -= s;                            // membrane subtract on spike
      // Output written once, never re-read: stream past L2.
      __builtin_nontemporal_store(s, (v4f*)(o + (long long)tt * kF));
    }
  };

  // Prologue: fill kNBuf-1 = 2 buffers (ASYNCcnt -> 32).
#pragma unroll
  for (int c = 0; c < kNBuf - 1; ++c) issue_chunk(c);

  // Steady state: after issuing chunk c+2 up to 48 ops outstanding;
  // asynccnt<=32 guarantees chunk c fully in LDS (async loads retire in order).
  for (int c = 0; c < kNChunk - (kNBuf - 1); ++c) {
    issue_chunk(c + kNBuf - 1);
    asm volatile("s_wait_asynccnt 32" ::: "memory");
    process_chunk(c);
  }
  // Tail peel: no more issues, wait thresholds shrink.
  asm volatile("s_wait_asynccnt 16" ::: "memory");
  process_chunk(kNChunk - 2);
  asm volatile("s_wait_asynccnt 0" ::: "memory");
  process_chunk(kNChunk - 1);
}

extern "C" void kernel_launch(void* const* d_in, const int* in_sizes, int n_in,
                              void* d_out, int out_size, void* d_ws, size_t ws_size,
                              hipStream_t stream) {
  const float* in  = (const float*)d_in[0];
  float*       out = (float*)d_out;

  const int total = in_sizes[0];            // B * T * F
  const int B     = total / (kT * kF);      // 16 for the reference shapes

  dim3 grid((unsigned)(B * kNFBlk));        // 16 * 16 = 256 blocks
  dim3 block((unsigned)kBlock);
  iaf_spike_kernel<<<grid, block, 0, stream>>>(in, out);
}